// Decoder_86938728006121
// MI455X (gfx1250) — compile-verified
//
#include <hip/hip_runtime.h>
#include <math.h>

// ---------------- problem constants ----------------
#define Bsz 128   // batch
#define Tt  400   // encoder length
#define KVd 128   // key/value dim
#define Vv  1024  // vocab
#define Ee  256   // embed dim
#define Hh  512   // decoder hidden
#define Ll  300   // decode length
#define SOS 1
#define NEG (-1e9f)

typedef float v2f __attribute__((ext_vector_type(2)));
typedef float v8f __attribute__((ext_vector_type(8)));

static __device__ __forceinline__ v8f wmma_f32(v2f a, v2f b, v8f c) {
  // D = A(16x4 f32) * B(4x16 f32) + C(16x16 f32)
  return __builtin_amdgcn_wmma_f32_16x16x4_f32(
      /*neg_a=*/false, a, /*neg_b=*/false, b,
      /*c_mod=*/(short)0, c, /*reuse_a=*/false, /*reuse_b=*/false);
}

static __device__ __forceinline__ float sigmoidf_(float x) {
  return 1.0f / (1.0f + __expf(-x));
}

// ---------------- workspace zeroing ----------------
__global__ void zero_kernel(float* p, int n) {
  int i = blockIdx.x * blockDim.x + threadIdx.x;
  if (i < n) p[i] = 0.0f;
}

// ---------------- LSTM cell 1 ----------------
// gates1(128 x 2048) = [emb_t, ctx](128 x 384) @ w_ih1^T + h1(128 x 512) @ w_hh1^T + b
// Block: 128 threads = 4 waves, wave w computes gate w's 16x16 tile. Fused LSTM update.
__global__ void lstm1_kernel(const int* __restrict__ y,
                             const float* __restrict__ emb,
                             const float* __restrict__ ctx,
                             const float* __restrict__ h1_cur,
                             float* __restrict__ h1_nxt,
                             float* __restrict__ c1,
                             const float* __restrict__ w_ih1,
                             const float* __restrict__ w_hh1,
                             const float* __restrict__ b_ih1,
                             const float* __restrict__ b_hh1,
                             int t) {
  __shared__ float gsh[4][16][16];
  const int wave = threadIdx.x >> 5;        // gate id 0..3 (i,f,g,o)
  const int lane = threadIdx.x & 31;
  const int half = lane >> 4;
  const int lr   = lane & 15;
  const int m0 = blockIdx.x * 16;           // batch tile
  const int j0 = blockIdx.y * 16;           // hidden-unit tile within gate
  const int mrow = m0 + lr;
  const int ncol = wave * Hh + j0 + lr;     // global gate column for this lane

  const int tok = (t == 0) ? SOS : y[mrow * Ll + (t - 1)];
  const float* erow = emb + (size_t)tok * Ee;
  const float* crow = ctx + (size_t)mrow * KVd;
  const float* wi   = w_ih1 + (size_t)ncol * (Ee + KVd);

  v8f acc = {0.f, 0.f, 0.f, 0.f, 0.f, 0.f, 0.f, 0.f};

  // x part: k in [0, 384): [emb | ctx]
#pragma unroll 4
  for (int k0 = 0; k0 < Ee + KVd; k0 += 4) {
    const int kb = k0 + 2 * half;
    v2f a, b;
    if (kb < Ee) { a.x = erow[kb];        a.y = erow[kb + 1]; }
    else         { a.x = crow[kb - Ee];   a.y = crow[kb - Ee + 1]; }
    b.x = wi[kb]; b.y = wi[kb + 1];
    acc = wmma_f32(a, b, acc);
  }
  // recurrent part: k in [0, 512)
  const float* hrow = h1_cur + (size_t)mrow * Hh;
  const float* wh   = w_hh1 + (size_t)ncol * Hh;
#pragma unroll 4
  for (int k0 = 0; k0 < Hh; k0 += 4) {
    const int kb = k0 + 2 * half;
    v2f a, b;
    a.x = hrow[kb]; a.y = hrow[kb + 1];
    b.x = wh[kb];   b.y = wh[kb + 1];
    acc = wmma_f32(a, b, acc);
  }

  const float bias = b_ih1[ncol] + b_hh1[ncol];
#pragma unroll
  for (int r = 0; r < 8; ++r)
    gsh[wave][r + 8 * half][lr] = acc[r] + bias;
  __syncthreads();

  // fused elementwise LSTM update (256 elements, 128 threads)
  for (int e = threadIdx.x; e < 256; e += 128) {
    const int ml = e >> 4, jl = e & 15;
    const float ig = gsh[0][ml][jl];
    const float fg = gsh[1][ml][jl];
    const float gg = gsh[2][ml][jl];
    const float og = gsh[3][ml][jl];
    const int gidx = (m0 + ml) * Hh + (j0 + jl);
    const float cold = c1[gidx];
    const float cn = sigmoidf_(fg) * cold + sigmoidf_(ig) * tanhf(gg);
    c1[gidx]     = cn;
    h1_nxt[gidx] = sigmoidf_(og) * tanhf(cn);
  }
}

// ---------------- LSTM cell 2 ----------------
// gates2(128 x 512) = h1(128 x 512) @ w_ih2^T + h2(128 x 128) @ w_hh2^T + b
__global__ void lstm2_kernel(const float* __restrict__ h1,
                             const float* __restrict__ h2_cur,
                             float* __restrict__ h2_nxt,
                             float* __restrict__ c2,
                             const float* __restrict__ w_ih2,
                             const float* __restrict__ w_hh2,
                             const float* __restrict__ b_ih2,
                             const float* __restrict__ b_hh2) {
  __shared__ float gsh[4][16][16];
  const int wave = threadIdx.x >> 5;
  const int lane = threadIdx.x & 31;
  const int half = lane >> 4;
  const int lr   = lane & 15;
  const int m0 = blockIdx.x * 16;
  const int j0 = blockIdx.y * 16;
  const int mrow = m0 + lr;
  const int ncol = wave * KVd + j0 + lr;

  v8f acc = {0.f, 0.f, 0.f, 0.f, 0.f, 0.f, 0.f, 0.f};

  const float* xrow = h1 + (size_t)mrow * Hh;
  const float* wi   = w_ih2 + (size_t)ncol * Hh;
#pragma unroll 4
  for (int k0 = 0; k0 < Hh; k0 += 4) {
    const int kb = k0 + 2 * half;
    v2f a, b;
    a.x = xrow[kb]; a.y = xrow[kb + 1];
    b.x = wi[kb];   b.y = wi[kb + 1];
    acc = wmma_f32(a, b, acc);
  }
  const float* hrow = h2_cur + (size_t)mrow * KVd;
  const float* wh   = w_hh2 + (size_t)ncol * KVd;
#pragma unroll 4
  for (int k0 = 0; k0 < KVd; k0 += 4) {
    const int kb = k0 + 2 * half;
    v2f a, b;
    a.x = hrow[kb]; a.y = hrow[kb + 1];
    b.x = wh[kb];   b.y = wh[kb + 1];
    acc = wmma_f32(a, b, acc);
  }

  const float bias = b_ih2[ncol] + b_hh2[ncol];
#pragma unroll
  for (int r = 0; r < 8; ++r)
    gsh[wave][r + 8 * half][lr] = acc[r] + bias;
  __syncthreads();

  for (int e = threadIdx.x; e < 256; e += 128) {
    const int ml = e >> 4, jl = e & 15;
    const float ig = gsh[0][ml][jl];
    const float fg = gsh[1][ml][jl];
    const float gg = gsh[2][ml][jl];
    const float og = gsh[3][ml][jl];
    const int gidx = (m0 + ml) * KVd + (j0 + jl);
    const float cold = c2[gidx];
    const float cn = sigmoidf_(fg) * cold + sigmoidf_(ig) * tanhf(gg);
    c2[gidx]     = cn;
    h2_nxt[gidx] = sigmoidf_(og) * tanhf(cn);
  }
}

// ---------------- attention (one block per batch row) ----------------
__global__ void attn_kernel(const float* __restrict__ enc_key,
                            const float* __restrict__ enc_value,
                            const int* __restrict__ enc_len,
                            const float* __restrict__ h2,    // query
                            float* __restrict__ ctx,
                            float* __restrict__ attn_plot,   // (L, T)
                            int t) {
  __shared__ float q[KVd];
  __shared__ float attnL[Tt];
  __shared__ float red[256];
  const int b   = blockIdx.x;
  const int tid = threadIdx.x;
  const int len = enc_len[b];

  if (tid < KVd) q[tid] = h2[b * KVd + tid];
  __syncthreads();

  // energies (each thread owns up to 2 encoder positions)
  float e0 = -INFINITY, e1 = -INFINITY;
  {
    const int t0 = tid;
    const float4* kp = (const float4*)(enc_key + ((size_t)b * Tt + t0) * KVd);
    const float4* qp = (const float4*)q;
    float s = 0.f;
#pragma unroll 8
    for (int d = 0; d < KVd / 4; ++d) {
      float4 kv = kp[d], qv = qp[d];
      s += kv.x * qv.x + kv.y * qv.y + kv.z * qv.z + kv.w * qv.w;
    }
    e0 = (t0 < len) ? s : NEG;
    attnL[t0] = e0;
  }
  if (tid + 256 < Tt) {
    const int t1 = tid + 256;
    const float4* kp = (const float4*)(enc_key + ((size_t)b * Tt + t1) * KVd);
    const float4* qp = (const float4*)q;
    float s = 0.f;
#pragma unroll 8
    for (int d = 0; d < KVd / 4; ++d) {
      float4 kv = kp[d], qv = qp[d];
      s += kv.x * qv.x + kv.y * qv.y + kv.z * qv.z + kv.w * qv.w;
    }
    e1 = (t1 < len) ? s : NEG;
    attnL[t1] = e1;
  }

  // row max
  red[tid] = fmaxf(e0, e1);
  __syncthreads();
  for (int s = 128; s > 0; s >>= 1) {
    if (tid < s) red[tid] = fmaxf(red[tid], red[tid + s]);
    __syncthreads();
  }
  const float mx = red[0];
  __syncthreads();

  // exp + sum
  float ls = 0.f;
  {
    const float ex = __expf(e0 - mx);
    attnL[tid] = ex; ls += ex;
  }
  if (tid + 256 < Tt) {
    const float ex = __expf(e1 - mx);
    attnL[tid + 256] = ex; ls += ex;
  }
  red[tid] = ls;
  __syncthreads();
  for (int s = 128; s > 0; s >>= 1) {
    if (tid < s) red[tid] += red[tid + s];
    __syncthreads();
  }
  const float inv = 1.0f / red[0];
  __syncthreads();

  // normalize (+ attn_plot for batch 0)
  attnL[tid] *= inv;
  if (tid + 256 < Tt) attnL[tid + 256] *= inv;
  if (b == 0) {
    attn_plot[(size_t)t * Tt + tid] = attnL[tid];
    if (tid + 256 < Tt) attn_plot[(size_t)t * Tt + tid + 256] = attnL[tid + 256];
  }
  __syncthreads();

  // context[d] = sum_t attn[t] * value[b][t][d]   (threads 0..127 own d)
  if (tid < KVd) {
    const float* vb = enc_value + (size_t)b * Tt * KVd + tid;
    float s = 0.f;
#pragma unroll 4
    for (int tt = 0; tt < Tt; ++tt) s += attnL[tt] * vb[(size_t)tt * KVd];
    ctx[b * KVd + tid] = s;
  }
}

// ---------------- output projection ----------------
// logits(128 x 1024) = [h2 | ctx](128 x 256) @ w_out^T + b_out -> predictions[:, t, :]
__global__ void out_kernel(const float* __restrict__ h2,
                           const float* __restrict__ ctx,
                           const float* __restrict__ w_out,
                           const float* __restrict__ b_out,
                           float* __restrict__ pred,
                           int t) {
  const int wave = threadIdx.x >> 5;        // 8 waves per block
  const int lane = threadIdx.x & 31;
  const int half = lane >> 4;
  const int lr   = lane & 15;
  const int m0 = blockIdx.x * 16;
  const int n0 = (blockIdx.y * 8 + wave) * 16;
  const int mrow = m0 + lr;
  const int ncol = n0 + lr;

  const float* hrow = h2  + (size_t)mrow * KVd;
  const float* crow = ctx + (size_t)mrow * KVd;
  const float* wrow = w_out + (size_t)ncol * (2 * KVd);

  v8f acc = {0.f, 0.f, 0.f, 0.f, 0.f, 0.f, 0.f, 0.f};
#pragma unroll 4
  for (int k0 = 0; k0 < 2 * KVd; k0 += 4) {
    const int kb = k0 + 2 * half;
    v2f a, b;
    if (kb < KVd) { a.x = hrow[kb];       a.y = hrow[kb + 1]; }
    else          { a.x = crow[kb - KVd]; a.y = crow[kb - KVd + 1]; }
    b.x = wrow[kb]; b.y = wrow[kb + 1];
    acc = wmma_f32(a, b, acc);
  }

  const float bias = b_out[ncol];
#pragma unroll
  for (int r = 0; r < 8; ++r) {
    const int m = m0 + r + 8 * half;
    pred[(size_t)m * Ll * Vv + (size_t)t * Vv + ncol] = acc[r] + bias;
  }
}

// ---------------- launcher ----------------
extern "C" void kernel_launch(void* const* d_in, const int* in_sizes, int n_in,
                              void* d_out, int out_size, void* d_ws, size_t ws_size,
                              hipStream_t stream) {
  const float* enc_key   = (const float*)d_in[0];
  const float* enc_value = (const float*)d_in[1];
  const int*   enc_len   = (const int*)d_in[2];
  const int*   y         = (const int*)d_in[3];
  const float* emb       = (const float*)d_in[4];
  const float* w_ih1     = (const float*)d_in[5];
  const float* w_hh1     = (const float*)d_in[6];
  const float* b_ih1     = (const float*)d_in[7];
  const float* b_hh1     = (const float*)d_in[8];
  const float* w_ih2     = (const float*)d_in[9];
  const float* w_hh2     = (const float*)d_in[10];
  const float* b_ih2     = (const float*)d_in[11];
  const float* b_hh2     = (const float*)d_in[12];
  const float* w_out     = (const float*)d_in[13];
  const float* b_out     = (const float*)d_in[14];

  float* pred      = (float*)d_out;                                   // (B, L, V)
  float* attn_plot = (float*)d_out + (size_t)Bsz * Ll * Vv;           // (L, T)

  // workspace layout (floats)
  float* ws  = (float*)d_ws;
  float* h1a = ws;                       // 128*512
  float* h1b = h1a + Bsz * Hh;           // 128*512
  float* c1  = h1b + Bsz * Hh;           // 128*512
  float* h2a = c1  + Bsz * Hh;           // 128*128
  float* h2b = h2a + Bsz * KVd;          // 128*128
  float* c2  = h2b + Bsz * KVd;          // 128*128
  float* ctx = c2  + Bsz * KVd;          // 128*128
  const int ws_floats = 3 * Bsz * Hh + 4 * Bsz * KVd;                 // 262144

  zero_kernel<<<(ws_floats + 255) / 256, 256, 0, stream>>>(ws, ws_floats);

  for (int t = 0; t < Ll; ++t) {
    const float* h1c = (t & 1) ? h1b : h1a;
    float*       h1n = (t & 1) ? h1a : h1b;
    const float* h2c = (t & 1) ? h2b : h2a;
    float*       h2n = (t & 1) ? h2a : h2b;

    lstm1_kernel<<<dim3(Bsz / 16, Hh / 16), 128, 0, stream>>>(
        y, emb, ctx, h1c, h1n, c1, w_ih1, w_hh1, b_ih1, b_hh1, t);

    lstm2_kernel<<<dim3(Bsz / 16, KVd / 16), 128, 0, stream>>>(
        h1n, h2c, h2n, c2, w_ih2, w_hh2, b_ih2, b_hh2);

    attn_kernel<<<Bsz, 256, 0, stream>>>(
        enc_key, enc_value, enc_len, h2n, ctx, attn_plot, t);

    out_kernel<<<dim3(Bsz / 16, Vv / 128), 256, 0, stream>>>(
        h2n, ctx, w_out, b_out, pred, t);
  }
}